// TransposeIrrepsLayout_63513976373465
// MI455X (gfx1250) — compile-verified
//
#include <hip/hip_runtime.h>
#include <stdint.h>

// Row layout: segments (u,v) = (128,1),(64,3),(32,5),(16,7)  -> 592 dwords/row
#define ROW_DW  592
#define ROW_V4  148               // float4 per row
#define RPB     8                 // rows per block
#define BLOCK   256               // 8 waves (wave32)
#define TILE_V4 (RPB * ROW_V4)    // 1184

typedef float v4f __attribute__((ext_vector_type(4)));
typedef __attribute__((address_space(3))) char lds_char;

// LDS byte offset of a __shared__ object (addrspace(3) pointers are LDS offsets)
__device__ __forceinline__ uint32_t lds_addr_of(void* p) {
    return (uint32_t)(uintptr_t)(lds_char*)p;
}

__global__ __launch_bounds__(BLOCK)
void transpose_irreps_kernel(const float* __restrict__ in,
                             float* __restrict__ out,
                             int nrows)
{
    __shared__ __align__(16) float smem[RPB * ROW_DW];

    const int t = threadIdx.x;
    const long long row0 = (long long)blockIdx.x * RPB;
    int rowsHere = nrows - (int)row0;
    if (rowsHere > RPB) rowsHere = RPB;
    const int nv4 = rowsHere * ROW_V4;

    const float* gin = in + row0 * (long long)ROW_DW;
    const uint32_t ldsBase = lds_addr_of((void*)smem);

    // ---- Phase 1: async coalesced copy global -> LDS, 16B per lane ----
    for (int i = t; i < nv4; i += BLOCK) {
        uint32_t lofs = ldsBase + (uint32_t)i * 16u;
        uint64_t gp   = (uint64_t)(uintptr_t)(gin + (size_t)i * 4);
        asm volatile("global_load_async_to_lds_b128 %0, %1, off"
                     :: "v"(lofs), "v"(gp) : "memory");
    }
    asm volatile("s_wait_asynccnt 0" ::: "memory");
    __syncthreads();

    // ---- Phase 2: permuted LDS reads, coalesced b128 NT stores ----
    float* gout = out + row0 * (long long)ROW_DW;
    for (int i = t; i < nv4; i += BLOCK) {
        const int row = i / ROW_V4;
        const int c4  = i - row * ROW_V4;
        const int c   = c4 * 4;           // output column (4-aligned)

        // out[off + j*u + i] = in[off + i*v + j]; group of 4 shares one j.
        int p, v;
        if (c < 128)      { p = c;                                          v = 1; }
        else if (c < 320) { int d = c - 128; p = 128 + (d & 63) * 3 + (d >> 6); v = 3; }
        else if (c < 480) { int d = c - 320; p = 320 + (d & 31) * 5 + (d >> 5); v = 5; }
        else              { int d = c - 480; p = 480 + (d & 15) * 7 + (d >> 4); v = 7; }

        const float* s = smem + row * ROW_DW + p;
        v4f o;
        o.x = s[0];
        o.y = s[v];
        o.z = s[2 * v];
        o.w = s[3 * v];

        __builtin_nontemporal_store(o, (v4f*)(gout + (size_t)row * ROW_DW + c));
    }
}

extern "C" void kernel_launch(void* const* d_in, const int* in_sizes, int n_in,
                              void* d_out, int out_size, void* d_ws, size_t ws_size,
                              hipStream_t stream) {
    const float* x = (const float*)d_in[0];
    float* y = (float*)d_out;
    const int nrows = in_sizes[0] / ROW_DW;   // 200000
    const int grid  = (nrows + RPB - 1) / RPB;
    transpose_irreps_kernel<<<grid, BLOCK, 0, stream>>>(x, y, nrows);
}